// ResidualMamba_65764539236888
// MI455X (gfx1250) — compile-verified
//
#include <hip/hip_runtime.h>
#include <hip/hip_bf16.h>
#include <math.h>

typedef __attribute__((ext_vector_type(2))) float v2f;
typedef __attribute__((ext_vector_type(8))) float v8f;

#define D_MODEL 512
#define D_STATE 16
#define D_CONVW 4
#define D_INNER 1024
#define DT_RANK 32
#define BATCH   4
#define SEQLEN  2048
#define ROWS    (BATCH * SEQLEN)          // 8192
#define XPROJ_N (DT_RANK + 2 * D_STATE)   // 64

#define BK      32                        // K tile
#define LDS_STR 36                        // padded row stride (floats): conflict-free + 16B aligned
#define TILE_F  (64 * LDS_STR)            // floats per matrix tile in LDS

// -------------------- helpers --------------------
__device__ __forceinline__ float wave_sum32(float v) {
  v += __shfl_xor(v, 16, 32);
  v += __shfl_xor(v, 8, 32);
  v += __shfl_xor(v, 4, 32);
  v += __shfl_xor(v, 2, 32);
  v += __shfl_xor(v, 1, 32);
  return v;
}

// -------------------- LayerNorm: one wave per row of 512 --------------------
__global__ void ln_kernel(const float* __restrict__ x,
                          const float* __restrict__ w,
                          const float* __restrict__ b,
                          float* __restrict__ xn) {
  int wave = (blockIdx.x * blockDim.x + threadIdx.x) >> 5;
  int lane = threadIdx.x & 31;
  if (wave >= ROWS) return;
  const float4* row = (const float4*)(x + (size_t)wave * D_MODEL);
  float4 v[4];
  float s = 0.f, s2 = 0.f;
#pragma unroll
  for (int i = 0; i < 4; i++) {
    float4 t = row[i * 32 + lane];
    v[i] = t;
    s  += t.x + t.y + t.z + t.w;
    s2 += t.x * t.x + t.y * t.y + t.z * t.z + t.w * t.w;
  }
  s  = wave_sum32(s);
  s2 = wave_sum32(s2);
  float mu  = s * (1.f / D_MODEL);
  float var = s2 * (1.f / D_MODEL) - mu * mu;
  float inv = rsqrtf(var + 1e-5f);
  float4* orow = (float4*)(xn + (size_t)wave * D_MODEL);
#pragma unroll
  for (int i = 0; i < 4; i++) {
    int idx = i * 32 + lane;
    float4 wv = ((const float4*)w)[idx];
    float4 bv = ((const float4*)b)[idx];
    float4 o;
    o.x = (v[i].x - mu) * inv * wv.x + bv.x;
    o.y = (v[i].y - mu) * inv * wv.y + bv.y;
    o.z = (v[i].z - mu) * inv * wv.z + bv.z;
    o.w = (v[i].w - mu) * inv * wv.w + bv.w;
    orow[idx] = o;
  }
}

// ---------- FP32 WMMA GEMM, LDS double-buffered: C[M,N] = A[M,K] * W[N,K]^T ----
// Block = 128 threads (4 waves) -> 64M x 64N macro-tile. K tiled by 32; each
// tile staged global->regs->LDS (padded stride 36: conflict-free ds_load_b64,
// aligned float4 stores); next tile's global loads overlap the 32-WMMA sweep
// out of the current LDS buffer. EPI: 0=store, 1=softplus(+bias),
// 2=gelu*mask+residual.
template <int EPI>
__global__ void wmma_gemm(const float* __restrict__ A, const float* __restrict__ W,
                          float* __restrict__ C, int M, int N, int K,
                          int lda, int ldw, int ldc,
                          const float* __restrict__ bias,
                          const float* __restrict__ resid,
                          const float* __restrict__ mask) {
  __shared__ float smem[2][2 * TILE_F];   // [buf][A tile | B tile]  = 36 KB

  int tid  = threadIdx.x;
  int wv   = tid >> 5;
  int lane = tid & 31;
  int hi   = lane >> 4;     // 0: K=0,1 ; 1: K=2,3
  int l16  = lane & 15;
  int n0b  = blockIdx.x * 64;
  int m0b  = blockIdx.y * 64;

  // cooperative-load decomposition: 512 float4 per matrix tile, 4 per thread
  int rr[4], cc[4];
#pragma unroll
  for (int p = 0; p < 4; ++p) {
    int lin = p * 128 + tid;
    rr[p] = lin >> 3;         // 0..63 (row in tile)
    cc[p] = (lin & 7) * 4;    // 0..28 (k offset in tile)
  }

  float4 ra[4], rb[4];
  // ---- prologue: tile 0 ----
#pragma unroll
  for (int p = 0; p < 4; ++p) {
    ra[p] = *(const float4*)(A + (size_t)(m0b + rr[p]) * lda + cc[p]);
    rb[p] = *(const float4*)(W + (size_t)(n0b + rr[p]) * ldw + cc[p]);
  }
#pragma unroll
  for (int p = 0; p < 4; ++p) {
    *(float4*)(&smem[0][0]      + rr[p] * LDS_STR + cc[p]) = ra[p];
    *(float4*)(&smem[0][TILE_F] + rr[p] * LDS_STR + cc[p]) = rb[p];
  }
  __syncthreads();

  v8f acc0 = {}, acc1 = {}, acc2 = {}, acc3 = {};
  int T = K / BK;
  for (int t = 0; t < T; ++t) {
    int cur = t & 1;
    bool pf = (t + 1 < T);
    if (pf) {
      int kt = (t + 1) * BK;
#pragma unroll
      for (int p = 0; p < 4; ++p) {
        ra[p] = *(const float4*)(A + (size_t)(m0b + rr[p]) * lda + kt + cc[p]);
        rb[p] = *(const float4*)(W + (size_t)(n0b + rr[p]) * ldw + kt + cc[p]);
      }
    }

    const float* Aw = &smem[cur][0]      + (wv * 16 + l16) * LDS_STR + hi * 2;
    const float* B0 = &smem[cur][TILE_F] + (l16 +  0) * LDS_STR + hi * 2;
    const float* B1 = &smem[cur][TILE_F] + (l16 + 16) * LDS_STR + hi * 2;
    const float* B2 = &smem[cur][TILE_F] + (l16 + 32) * LDS_STR + hi * 2;
    const float* B3 = &smem[cur][TILE_F] + (l16 + 48) * LDS_STR + hi * 2;
#pragma unroll
    for (int kk = 0; kk < BK; kk += 4) {
      v2f a  = *(const v2f*)(Aw + kk);
      v2f b0 = *(const v2f*)(B0 + kk);
      v2f b1 = *(const v2f*)(B1 + kk);
      v2f b2 = *(const v2f*)(B2 + kk);
      v2f b3 = *(const v2f*)(B3 + kk);
      acc0 = __builtin_amdgcn_wmma_f32_16x16x4_f32(false, a, false, b0, (short)0, acc0, false, false);
      acc1 = __builtin_amdgcn_wmma_f32_16x16x4_f32(false, a, false, b1, (short)0, acc1, false, false);
      acc2 = __builtin_amdgcn_wmma_f32_16x16x4_f32(false, a, false, b2, (short)0, acc2, false, false);
      acc3 = __builtin_amdgcn_wmma_f32_16x16x4_f32(false, a, false, b3, (short)0, acc3, false, false);
    }

    if (pf) {
      int nxt = (t + 1) & 1;
#pragma unroll
      for (int p = 0; p < 4; ++p) {
        *(float4*)(&smem[nxt][0]      + rr[p] * LDS_STR + cc[p]) = ra[p];
        *(float4*)(&smem[nxt][TILE_F] + rr[p] * LDS_STR + cc[p]) = rb[p];
      }
      __syncthreads();
    }
  }

  // ---- epilogue ----
  int m0 = m0b + wv * 16;
  v8f accs[4] = {acc0, acc1, acc2, acc3};
#pragma unroll
  for (int tt = 0; tt < 4; tt++) {
    int col = n0b + tt * 16 + l16;
#pragma unroll
    for (int r = 0; r < 8; r++) {
      int row = m0 + hi * 8 + r;
      float val = accs[tt][r];
      if (EPI == 1) {
        float s = val + bias[col];
        val = (s > 20.f) ? s : log1pf(__expf(s));       // softplus
      } else if (EPI == 2) {
        float g = 0.5f * val * (1.f + erff(val * 0.70710678118654752f)); // exact gelu
        val = g * mask[row] + resid[(size_t)row * ldc + col];
      }
      C[(size_t)row * ldc + col] = val;
    }
  }
}

// -------------------- depthwise causal conv1d (k=4) + SiLU --------------------
__global__ void conv_silu_kernel(const float* __restrict__ xz,
                                 const float* __restrict__ cw,
                                 const float* __restrict__ cb,
                                 float* __restrict__ xm) {
  size_t idx = (size_t)blockIdx.x * blockDim.x + threadIdx.x;
  if (idx >= (size_t)ROWS * D_INNER) return;
  int d   = (int)(idx & (D_INNER - 1));
  int row = (int)(idx >> 10);
  int l   = row & (SEQLEN - 1);
  float acc = cb[d];
#pragma unroll
  for (int j = 0; j < D_CONVW; j++) {
    int ls = l - (D_CONVW - 1) + j;
    if (ls >= 0) {
      acc += cw[d * D_CONVW + j] * xz[(size_t)(row - (D_CONVW - 1) + j) * (2 * D_INNER) + d];
    }
  }
  xm[idx] = acc * (1.f / (1.f + __expf(-acc)));   // silu
}

// -------------------- selective scan: wave = 2 channels x 16 states --------------------
// y (written in place over xm) = (scan_out + u*D) * silu(z)
__global__ void scan_kernel(const float* __restrict__ delta,
                            float* __restrict__ xm,          // in: u, out: y
                            const float* __restrict__ xdbl,  // [rows,64]: dt|B|C
                            const float* __restrict__ xz,    // z at cols [1024,2048)
                            const float* __restrict__ A_log,
                            const float* __restrict__ Dp) {
  int g    = (blockIdx.x * blockDim.x + threadIdx.x) >> 5;
  int lane = threadIdx.x & 31;
  int n    = lane & 15;
  int dl   = lane >> 4;
  int b    = g / (D_INNER / 2);
  int dpr  = g % (D_INNER / 2);
  if (b >= BATCH) return;
  int d = dpr * 2 + dl;
  float An = -__expf(A_log[d * D_STATE + n]);
  float Dd = Dp[d];
  float h = 0.f;
  for (int l = 0; l < SEQLEN; l++) {
    size_t row = (size_t)b * SEQLEN + l;
    float dlt = delta[row * D_INNER + d];
    float u   = xm[row * D_INNER + d];
    float Bt  = xdbl[row * XPROJ_N + DT_RANK + n];
    float Ct  = xdbl[row * XPROJ_N + DT_RANK + D_STATE + n];
    h = __expf(dlt * An) * h + (dlt * u) * Bt;
    float p = h * Ct;
    p += __shfl_xor(p, 1, 32);
    p += __shfl_xor(p, 2, 32);
    p += __shfl_xor(p, 4, 32);
    p += __shfl_xor(p, 8, 32);
    if (n == 0) {
      float yv = p + u * Dd;
      float z  = xz[row * (2 * D_INNER) + D_INNER + d];
      yv *= z * (1.f / (1.f + __expf(-z)));            // * silu(z)
      xm[row * D_INNER + d] = yv;
    }
  }
}

// -------------------- copy mask to second tuple output --------------------
__global__ void mask_copy_kernel(const float* __restrict__ mask, float* __restrict__ out2) {
  int i = blockIdx.x * blockDim.x + threadIdx.x;
  if (i < ROWS) out2[i] = mask[i];
}

// -------------------- launcher --------------------
extern "C" void kernel_launch(void* const* d_in, const int* in_sizes, int n_in,
                              void* d_out, int out_size, void* d_ws, size_t ws_size,
                              hipStream_t stream) {
  const float* x      = (const float*)d_in[0];
  const float* mask   = (const float*)d_in[1];
  const float* ln_w   = (const float*)d_in[2];
  const float* ln_b   = (const float*)d_in[3];
  const float* W_in   = (const float*)d_in[4];
  const float* conv_w = (const float*)d_in[5];
  const float* conv_b = (const float*)d_in[6];
  const float* W_xprj = (const float*)d_in[7];
  const float* W_dt   = (const float*)d_in[8];
  const float* b_dt   = (const float*)d_in[9];
  const float* A_log  = (const float*)d_in[10];
  const float* D_par  = (const float*)d_in[11];
  const float* W_out  = (const float*)d_in[12];
  float* out = (float*)d_out;

  float* ws    = (float*)d_ws;
  float* xn    = ws;                                   // 8192*512
  float* xz    = xn   + (size_t)ROWS * D_MODEL;        // 8192*2048
  float* xm    = xz   + (size_t)ROWS * 2 * D_INNER;    // 8192*1024 (u, then y in place)
  float* xdbl  = xm   + (size_t)ROWS * D_INNER;        // 8192*64
  float* delta = xdbl + (size_t)ROWS * XPROJ_N;        // 8192*1024

  // 1) LayerNorm
  ln_kernel<<<ROWS / 8, 256, 0, stream>>>(x, ln_w, ln_b, xn);

  // 2) xz = xn @ W_in^T   (M=8192,N=2048,K=512)
  {
    dim3 g(2 * D_INNER / 64, ROWS / 64);
    wmma_gemm<0><<<g, 128, 0, stream>>>(xn, W_in, xz, ROWS, 2 * D_INNER, D_MODEL,
                                        D_MODEL, D_MODEL, 2 * D_INNER,
                                        nullptr, nullptr, nullptr);
  }

  // 3) depthwise conv + silu -> xm
  conv_silu_kernel<<<(ROWS * D_INNER) / 256, 256, 0, stream>>>(xz, conv_w, conv_b, xm);

  // 4) x_dbl = xm @ W_xproj^T  (M=8192,N=64,K=1024)
  {
    dim3 g(XPROJ_N / 64, ROWS / 64);
    wmma_gemm<0><<<g, 128, 0, stream>>>(xm, W_xprj, xdbl, ROWS, XPROJ_N, D_INNER,
                                        D_INNER, D_INNER, XPROJ_N,
                                        nullptr, nullptr, nullptr);
  }

  // 5) delta = softplus(dt @ W_dt^T + b_dt)  (M=8192,N=1024,K=32, A has lda=64)
  {
    dim3 g(D_INNER / 64, ROWS / 64);
    wmma_gemm<1><<<g, 128, 0, stream>>>(xdbl, W_dt, delta, ROWS, D_INNER, DT_RANK,
                                        XPROJ_N, DT_RANK, D_INNER,
                                        b_dt, nullptr, nullptr);
  }

  // 6) selective scan (+ *D, *silu(z)) in place over xm
  {
    int waves = BATCH * (D_INNER / 2);   // 2048 waves
    scan_kernel<<<(waves * 32) / 256, 256, 0, stream>>>(delta, xm, xdbl, xz,
                                                        A_log, D_par);
  }

  // 7) out = gelu(y @ W_out^T) * mask + residual  (M=8192,N=512,K=1024)
  {
    dim3 g(D_MODEL / 64, ROWS / 64);
    wmma_gemm<2><<<g, 128, 0, stream>>>(xm, W_out, out, ROWS, D_MODEL, D_INNER,
                                        D_INNER, D_INNER, D_MODEL,
                                        nullptr, x, mask);
  }

  // 8) second tuple output: mask passthrough
  mask_copy_kernel<<<ROWS / 256, 256, 0, stream>>>(mask, out + (size_t)ROWS * D_MODEL);
}